// ExemplarNoAttention_3624952398348
// MI455X (gfx1250) — compile-verified
//
#include <hip/hip_runtime.h>
#include <hip/hip_bf16.h>

#define NUM_CLASSES 10
#define ECH 32          // exemplar chunks per class segment (parallelism over Ne)
#define SORT_T 256

typedef __attribute__((ext_vector_type(2))) float v2f;
typedef __attribute__((ext_vector_type(8))) float v8f;

// ---------------------------------------------------------------------------
// Kernel 1: deterministic stable counting sort of exemplar indices by class.
// Single workgroup; thread t owns a contiguous slice, serial prefix by t0.
// ---------------------------------------------------------------------------
__global__ void sort_labels_kernel(const int* __restrict__ labels, int n,
                                   int* __restrict__ perm, int* __restrict__ offs) {
    __shared__ int counts[SORT_T][NUM_CLASSES];   // 10 KB LDS
    const int t = threadIdx.x;
    const int per = (n + SORT_T - 1) / SORT_T;
    const int lo = t * per;
    const int hi = min(lo + per, n);

    int c[NUM_CLASSES];
#pragma unroll
    for (int k = 0; k < NUM_CLASSES; ++k) c[k] = 0;
    for (int i = lo; i < hi; ++i) {
        int l = labels[i];
        l = (l < 0) ? 0 : (l >= NUM_CLASSES ? NUM_CLASSES - 1 : l);
        c[l]++;
    }
#pragma unroll
    for (int k = 0; k < NUM_CLASSES; ++k) counts[t][k] = c[k];
    __syncthreads();

    if (t == 0) {
        int run = 0;
        for (int k = 0; k < NUM_CLASSES; ++k) {
            offs[k] = run;
            for (int tt = 0; tt < SORT_T; ++tt) {
                int v = counts[tt][k];
                counts[tt][k] = run;
                run += v;
            }
        }
        offs[NUM_CLASSES] = run;
    }
    __syncthreads();

    int pos[NUM_CLASSES];
#pragma unroll
    for (int k = 0; k < NUM_CLASSES; ++k) pos[k] = counts[t][k];
    for (int i = lo; i < hi; ++i) {
        int l = labels[i];
        l = (l < 0) ? 0 : (l >= NUM_CLASSES ? NUM_CLASSES - 1 : l);
        perm[pos[l]++] = i;
    }
}

// ---------------------------------------------------------------------------
// Kernel 2: gathered squared norms of exemplar rows in sorted order.
// ---------------------------------------------------------------------------
__global__ void e2_gather_kernel(const float* __restrict__ ex,
                                 const int* __restrict__ perm,
                                 float* __restrict__ e2s, int n) {
    int i = blockIdx.x * blockDim.x + threadIdx.x;
    if (i >= n) return;
    const float4* p = (const float4*)(ex + (size_t)perm[i] * 64);
    float s = 0.f;
#pragma unroll
    for (int j = 0; j < 16; ++j) {
        float4 v = p[j];
        s += v.x * v.x + v.y * v.y + v.z * v.z + v.w * v.w;
    }
    e2s[i] = s;
}

// ---------------------------------------------------------------------------
// Kernel 3: main WMMA kernel. One wave per (16-row batch tile, chunk).
// D = x_tile (16xK=64, f32) @ ex_tile^T via 16x V_WMMA_F32_16X16X4_F32,
// split into two independent K-chains for ILP. Classes are contiguous in
// sorted order, so binning is just a running per-class register sum.
// ---------------------------------------------------------------------------
__launch_bounds__(32)
__global__ void exemplar_wmma_kernel(const float* __restrict__ x,
                                     const float* __restrict__ ex,
                                     const float* __restrict__ beta_raw,
                                     const int* __restrict__ perm,
                                     const int* __restrict__ offs,
                                     const float* __restrict__ e2s,
                                     float* __restrict__ partials,
                                     int batch) {
    const int lane = threadIdx.x;      // 0..31 (wave32)
    const int h    = lane >> 4;        // lane half
    const int n    = lane & 15;        // tile column / A-row
    const int tile  = blockIdx.x;      // batch tile (16 rows)
    const int chunk = blockIdx.y;      // exemplar chunk

    // --- Preload A: x tile, lane holds row m = n, K-pairs {4s+2h, 4s+2h+1}.
    const float* xr = x + (size_t)(tile * 16 + n) * 64;
    v2f A[16];
#pragma unroll
    for (int s = 0; s < 16; ++s) {
        float2 v = *(const float2*)(xr + 4 * s + 2 * h);
        A[s].x = v.x;
        A[s].y = v.y;
    }

    // --- ||x_m||^2 for the 8 D-rows this lane owns: m = r + 8h.
    float x2v[8];
#pragma unroll
    for (int r = 0; r < 8; ++r) {
        const float4* p = (const float4*)(x + (size_t)(tile * 16 + r + 8 * h) * 64);
        float s = 0.f;
#pragma unroll
        for (int j = 0; j < 16; ++j) {
            float4 v = p[j];
            s += v.x * v.x + v.y * v.y + v.z * v.z + v.w * v.w;
        }
        x2v[r] = s;
    }

    // softplus(beta_raw)
    const float beta = logf(1.0f + __expf(beta_raw[0]));

    for (int c = 0; c < NUM_CLASSES; ++c) {
        const int seg_lo = offs[c];
        const int seg_hi = offs[c + 1];
        const int len    = seg_hi - seg_lo;
        const int per    = (len + ECH - 1) / ECH;
        const int lo     = seg_lo + chunk * per;
        const int hi     = min(lo + per, seg_hi);

        float acc[8];
#pragma unroll
        for (int r = 0; r < 8; ++r) acc[r] = 0.f;

        for (int t = lo; t < hi; t += 16) {
            const int  e     = t + n;                 // sorted position, this column
            const bool valid = (e < hi);
            const int  ec    = valid ? e : (hi - 1);  // safe clamp for loads
            const float* er  = ex + (size_t)perm[ec] * 64;

            v8f D0 = {};
            v8f D1 = {};
#pragma unroll
            for (int s = 0; s < 8; ++s) {
                float2 bv = *(const float2*)(er + 4 * s + 2 * h);
                v2f b; b.x = bv.x; b.y = bv.y;
                D0 = __builtin_amdgcn_wmma_f32_16x16x4_f32(
                        false, A[s], false, b, (short)0, D0, false, false);
            }
#pragma unroll
            for (int s = 8; s < 16; ++s) {
                float2 bv = *(const float2*)(er + 4 * s + 2 * h);
                v2f b; b.x = bv.x; b.y = bv.y;
                D1 = __builtin_amdgcn_wmma_f32_16x16x4_f32(
                        false, A[s], false, b, (short)0, D1, false, false);
            }

            const float e2 = e2s[ec];
            const float m  = valid ? 1.f : 0.f;
#pragma unroll
            for (int r = 0; r < 8; ++r) {
                float dot = D0[r] + D1[r];
                float d2  = fmaf(-2.f, dot, x2v[r] + e2);
                d2 = fmaxf(d2, 0.f);
                acc[r] = fmaf(m, __expf(-beta * d2), acc[r]);
            }
        }

        // Reduce across the 16 tile columns (within each 16-lane half).
#pragma unroll
        for (int r = 0; r < 8; ++r) {
            float v = acc[r];
            v += __shfl_xor(v, 1, 32);
            v += __shfl_xor(v, 2, 32);
            v += __shfl_xor(v, 4, 32);
            v += __shfl_xor(v, 8, 32);
            acc[r] = v;
        }
        if (n == 0) {  // lanes 0 and 16 write rows r and r+8 respectively
#pragma unroll
            for (int r = 0; r < 8; ++r) {
                const int row = tile * 16 + r + 8 * h;
                partials[((size_t)chunk * batch + row) * NUM_CLASSES + c] = acc[r];
            }
        }
    }
}

// ---------------------------------------------------------------------------
// Kernel 4: deterministic fixed-order reduction over chunks + log.
// ---------------------------------------------------------------------------
__global__ void reduce_log_kernel(const float* __restrict__ partials,
                                  float* __restrict__ out, int batch) {
    const int i = blockIdx.x * blockDim.x + threadIdx.x;
    const int total = batch * NUM_CLASSES;
    if (i >= total) return;
    float s = 0.f;
    for (int ch = 0; ch < ECH; ++ch)
        s += partials[(size_t)ch * total + i];
    out[i] = logf(s + 1e-12f);   // GAMMA == 1.0
}

// ---------------------------------------------------------------------------
extern "C" void kernel_launch(void* const* d_in, const int* in_sizes, int n_in,
                              void* d_out, int out_size, void* d_ws, size_t ws_size,
                              hipStream_t stream) {
    const float* x        = (const float*)d_in[0];
    const float* ex       = (const float*)d_in[1];
    const int*   labels   = (const int*)d_in[2];
    const float* beta_raw = (const float*)d_in[3];
    float*       out      = (float*)d_out;

    const int batch = in_sizes[0] / 64;   // 1024
    const int ne    = in_sizes[2];        // 50000

    char* ws = (char*)d_ws;
    size_t off = 0;
    int* perm = (int*)(ws + off);
    off += (((size_t)ne * 4) + 255) & ~(size_t)255;
    int* offs = (int*)(ws + off);
    off += 256;
    float* e2s = (float*)(ws + off);
    off += (((size_t)ne * 4) + 255) & ~(size_t)255;
    float* partials = (float*)(ws + off);   // ECH * batch * 10 floats (~1.3 MB)

    sort_labels_kernel<<<1, SORT_T, 0, stream>>>(labels, ne, perm, offs);
    e2_gather_kernel<<<(ne + 255) / 256, 256, 0, stream>>>(ex, perm, e2s, ne);

    dim3 grid(batch / 16, ECH);
    exemplar_wmma_kernel<<<grid, 32, 0, stream>>>(x, ex, beta_raw, perm, offs,
                                                  e2s, partials, batch);

    const int total = batch * NUM_CLASSES;
    reduce_log_kernel<<<(total + 255) / 256, 256, 0, stream>>>(partials, out, batch);
}